// SuperResolution_89876485636254
// MI455X (gfx1250) — compile-verified
//
#include <hip/hip_runtime.h>
#include <stdint.h>

#define IMG      28
#define IMG2     784          // 28*28
#define GPI      196          // 4-pixel groups per image (784/4)
#define GPR      7            // groups per row (28/4)
#define CENTER   14
#define IPB      8            // images per block (8*784*4B = 25088 B LDS)
#define THREADS  256          // 8 wave32

typedef float v4f __attribute__((ext_vector_type(4)));

// CDNA5 async global->LDS 128-bit copy (ASYNCcnt-tracked). Inline asm is
// portable across ROCm-7.2 / amdgpu-toolchain builtin-arity differences.
// Operands: VDST = VGPR with LDS byte address, VADDR = 32-bit byte offset,
// SADDR = SGPR pair base (GVS mode).
__device__ __forceinline__ void async_load_b128(uint32_t lds_addr,
                                                const float* sbase,
                                                uint32_t voff) {
  asm volatile("global_load_async_to_lds_b128 %0, %1, %2"
               :
               : "v"(lds_addr), "v"(voff), "s"(sbase)
               : "memory");
}

__device__ __forceinline__ void wait_async0() {
  asm volatile("s_wait_asynccnt 0" ::: "memory");
}

__global__ __launch_bounds__(THREADS, 2)
void Diffusion_restore_step_kernel(const float* __restrict__ x,
                                   const int*   __restrict__ t,
                                   const float* __restrict__ Wk,
                                   const float* __restrict__ bias,
                                   float* __restrict__ out,
                                   int nimg) {
  __shared__ __align__(16) float smem[IPB * IMG2];

  const int img0 = blockIdx.x * IPB;
  if (img0 >= nimg) return;
  const int imgs   = min(IPB, nimg - img0);
  const int npix   = imgs * IMG2;
  const int nchunk = npix >> 2;               // 16-byte chunks

  const float* gsrc = x + (size_t)img0 * IMG2;          // block-uniform -> SGPR pair
  const uint32_t lds0 = (uint32_t)(uintptr_t)&smem[0];  // addr[31:0] == LDS byte addr

  // ---- Phase 1: stage images into LDS via async B128 transfers ----
  for (int ch = threadIdx.x; ch < nchunk; ch += THREADS) {
    const uint32_t boff = (uint32_t)ch << 4;
    async_load_b128(lds0 + boff, gsrc, boff);
  }
  wait_async0();        // this wave's async copies landed in LDS
  __syncthreads();      // all waves' copies visible

  // Block-uniform weights -> scalar loads, live in registers.
  float w[9];
#pragma unroll
  for (int i = 0; i < 9; ++i) w[i] = Wk[i];
  const float bb = bias[0];

  float* gdst = out + (size_t)img0 * IMG2;

  // ---- Phase 2: 4 pixels/thread. 3x b128 + 6x b32 LDS reads per group,
  //      9 FMAs/pixel (irreducible), one b128 NT store per group. ----
  const int ngroup = npix >> 2;
  for (int g = threadIdx.x; g < ngroup; g += THREADS) {
    const int img = g / GPI;
    const int rem = g - img * GPI;
    const int r   = rem / GPR;
    const int c4  = (rem - r * GPR) << 2;     // 0,4,...,24
    const float* s = &smem[img * IMG2 + r * IMG];

    // Row windows [c4-1 .. c4+4] as 6-wide register arrays (zero padded).
    float T[6] = {0.f, 0.f, 0.f, 0.f, 0.f, 0.f};
    float M[6];
    float B[6] = {0.f, 0.f, 0.f, 0.f, 0.f, 0.f};

    {
      const v4f mv = *(const v4f*)(s + c4);   // 16B-aligned ds_load_b128
      M[0] = (c4 > 0)  ? s[c4 - 1] : 0.f;
      M[1] = mv.x; M[2] = mv.y; M[3] = mv.z; M[4] = mv.w;
      M[5] = (c4 < 24) ? s[c4 + 4] : 0.f;
    }
    if (r > 0) {
      const float* st = s - IMG;
      const v4f tv = *(const v4f*)(st + c4);
      T[0] = (c4 > 0)  ? st[c4 - 1] : 0.f;
      T[1] = tv.x; T[2] = tv.y; T[3] = tv.z; T[4] = tv.w;
      T[5] = (c4 < 24) ? st[c4 + 4] : 0.f;
    }
    if (r < IMG - 1) {
      const float* sb = s + IMG;
      const v4f bv = *(const v4f*)(sb + c4);
      B[0] = (c4 > 0)  ? sb[c4 - 1] : 0.f;
      B[1] = bv.x; B[2] = bv.y; B[3] = bv.z; B[4] = bv.w;
      B[5] = (c4 < 24) ? sb[c4 + 4] : 0.f;
    }

    // Per-group scalars for the annulus test (amortized over 4 pixels).
    const int ti   = t[img0 + img];
    const int tm   = ti - 1;
    const int dy   = r - CENTER;
    const int dy2  = dy * dy;
    const int ti2  = ti * ti;
    const int tm2  = tm * tm;
    const bool tok  = (ti >= 0);
    const bool tmok = (tm >= 0);

    float res[4];
#pragma unroll
    for (int j = 0; j < 4; ++j) {
      float acc = bb;
      acc = fmaf(w[0], T[j],     acc);
      acc = fmaf(w[1], T[j + 1], acc);
      acc = fmaf(w[2], T[j + 2], acc);
      acc = fmaf(w[3], M[j],     acc);
      acc = fmaf(w[4], M[j + 1], acc);
      acc = fmaf(w[5], M[j + 2], acc);
      acc = fmaf(w[6], B[j],     acc);
      acc = fmaf(w[7], B[j + 1], acc);
      acc = fmaf(w[8], B[j + 2], acc);

      const int dx = c4 + j - CENTER;
      const int d2 = dy2 + dx * dx;
      const bool in_t  = tok  && (d2 <= ti2);
      const bool in_tm = tmok && (d2 <= tm2);

      // Literal reference order: (x - fp_t) + fp_{t-1}
      const float fpt  = in_t  ? 0.f : acc;
      const float fptm = in_tm ? 0.f : acc;
      res[j] = (M[j + 1] - fpt) + fptm;
    }

    const v4f rv = {res[0], res[1], res[2], res[3]};
    __builtin_nontemporal_store(rv, (v4f*)(gdst + (size_t)g * 4));  // 16B aligned
  }
}

extern "C" void kernel_launch(void* const* d_in, const int* in_sizes, int n_in,
                              void* d_out, int out_size, void* d_ws, size_t ws_size,
                              hipStream_t stream) {
  (void)n_in; (void)out_size; (void)d_ws; (void)ws_size;
  const float* x    = (const float*)d_in[0];
  const int*   t    = (const int*)  d_in[1];
  const float* Wk   = (const float*)d_in[2];
  const float* bias = (const float*)d_in[3];
  float*       outp = (float*)d_out;

  const int nimg   = in_sizes[0] / IMG2;              // 65536
  const int blocks = (nimg + IPB - 1) / IPB;          // 8192
  Diffusion_restore_step_kernel<<<dim3(blocks), dim3(THREADS), 0, stream>>>(
      x, t, Wk, bias, outp, nimg);
}